// CAREConv_4672924418438
// MI455X (gfx1250) — compile-verified
//
#include <hip/hip_runtime.h>
#include <cmath>

typedef float v2f __attribute__((ext_vector_type(2)));
typedef float v8f __attribute__((ext_vector_type(8)));

#define N_NODES 50000
#define N_EDGES 800000
#define IN_DIM  128
#define OUT_DIM 128
#define N_CLS   40
#define N_CLS_PAD 48

// ---------------------------------------------------------------------------
// Kernel 0: zero-pad W_mlp [128,40] -> Wpad [128,48] so the WMMA loop needs
// no predicated loads (keeps EXEC all-ones, no s_and_saveexec churn).
// ---------------------------------------------------------------------------
__global__ void padW_kernel(const float* __restrict__ Wm, float* __restrict__ Wpad) {
  const int i = blockIdx.x * blockDim.x + threadIdx.x;
  if (i < IN_DIM * N_CLS_PAD) {
    const int r = i / N_CLS_PAD, c = i % N_CLS_PAD;
    Wpad[i] = (c < N_CLS) ? Wm[r * N_CLS + c] : 0.0f;
  }
}

// ---------------------------------------------------------------------------
// Kernel 1: t = tanh(feat @ W_mlp + b_mlp)   [50000,128]x[128,40] -> [50000,40]
// WMMA f32 16x16x4. One wave per 16-row block, 3 column tiles of the padded
// 48-col weight. 3125*16 == 50000 exactly -> EXEC all ones, no guards.
// A layout (16x4 f32): lanes 0-15 hold M, VGPR0=K0/K2, VGPR1=K1/K3 by half-wave.
// B layout (4x16 f32): lanes hold N, VGPR0=K0/K2, VGPR1=K1/K3 by half-wave.
// C/D layout: value (v, lane) -> M = v + 8*(lane>>4), N = lane&15.
// ---------------------------------------------------------------------------
__global__ __launch_bounds__(32) void mlp_tanh_kernel(
    const float* __restrict__ feat, const float* __restrict__ Wpad,
    const float* __restrict__ bm, float* __restrict__ t) {
  const int lane = threadIdx.x & 31;
  const int hf   = lane >> 4;
  const int m    = lane & 15;
  const int row0 = blockIdx.x * 16;

  v8f acc[3] = {};
  const float* arow = feat + (size_t)(row0 + m) * IN_DIM;

#pragma unroll 4
  for (int k = 0; k < IN_DIM; k += 4) {
    const int ka = k + 2 * hf;
    v2f a;
    a.x = arow[ka];
    a.y = arow[ka + 1];
#pragma unroll
    for (int tile = 0; tile < 3; ++tile) {
      const int col = tile * 16 + m;
      v2f b;
      b.x = Wpad[ka * N_CLS_PAD + col];
      b.y = Wpad[(ka + 1) * N_CLS_PAD + col];
      acc[tile] = __builtin_amdgcn_wmma_f32_16x16x4_f32(
          false, a, false, b, (short)0, acc[tile], false, false);
    }
  }

#pragma unroll
  for (int tile = 0; tile < 3; ++tile) {
    const int col = tile * 16 + m;
    if (col < N_CLS) {
      const float bias = bm[col];
#pragma unroll
      for (int v = 0; v < 8; ++v) {
        const int r = row0 + v + 8 * hf;
        t[(size_t)r * N_CLS + col] = tanhf(acc[tile][v] + bias);
      }
    }
  }
}

// ---------------------------------------------------------------------------
// Kernel 2: dist[e] = || t[src[e]] - t[dst[e]] ||_1   (40 classes, float4 x10)
// t rows are 160 B -> float4 aligned; t (8 MB) is L2-resident.
// ---------------------------------------------------------------------------
__global__ void dist_kernel(const int* __restrict__ src,
                            const int* __restrict__ dst,
                            const float* __restrict__ t,
                            float* __restrict__ dist) {
  const int e = blockIdx.x * blockDim.x + threadIdx.x;
  if (e >= N_EDGES) return;
  const float4* ts = (const float4*)(t + (size_t)src[e] * N_CLS);
  const float4* td = (const float4*)(t + (size_t)dst[e] * N_CLS);
  float s = 0.0f;
#pragma unroll
  for (int i = 0; i < 10; ++i) {
    const float4 a = ts[i], b = td[i];
    s += fabsf(a.x - b.x) + fabsf(a.y - b.y) + fabsf(a.z - b.z) + fabsf(a.w - b.w);
  }
  dist[e] = s;
}

// ---------------------------------------------------------------------------
// Kernel 3: in-degree histogram
// ---------------------------------------------------------------------------
__global__ void degree_kernel(const int* __restrict__ dst, int* __restrict__ deg) {
  const int e = blockIdx.x * blockDim.x + threadIdx.x;
  if (e < N_EDGES) atomicAdd(&deg[dst[e]], 1);
}

// ---------------------------------------------------------------------------
// Kernel 4: exclusive scan of deg -> seg_start. Single 1024-thread block,
// chunked Hillis-Steele (49 chunks). 50000 ints: trivial.
// ---------------------------------------------------------------------------
__global__ __launch_bounds__(1024) void scan_kernel(const int* __restrict__ deg,
                                                    int* __restrict__ seg_start,
                                                    int n) {
  __shared__ int sh[1024];
  __shared__ int base_s;
  if (threadIdx.x == 0) base_s = 0;
  __syncthreads();
  for (int start = 0; start < n; start += 1024) {
    const int i = start + (int)threadIdx.x;
    const int v = (i < n) ? deg[i] : 0;
    sh[threadIdx.x] = v;
    __syncthreads();
    for (int off = 1; off < 1024; off <<= 1) {
      const int tv = (threadIdx.x >= (unsigned)off) ? sh[threadIdx.x - off] : 0;
      __syncthreads();
      sh[threadIdx.x] += tv;
      __syncthreads();
    }
    const int incl = sh[threadIdx.x];
    if (i < n) seg_start[i] = base_s + incl - v;
    __syncthreads();
    if (threadIdx.x == 0) base_s += sh[1023];
    __syncthreads();
  }
}

// ---------------------------------------------------------------------------
// Kernel 5: counting-sort scatter: group edges by dst (order inside a segment
// is atomic-order, i.e. nondeterministic, but the later rank comparator makes
// the final selection deterministic).
// ---------------------------------------------------------------------------
__global__ void scatter_kernel(const int* __restrict__ dst,
                               const float* __restrict__ dist,
                               const int* __restrict__ seg_start,
                               int* __restrict__ cursor,
                               int* __restrict__ eid_sorted,
                               float* __restrict__ dist_sorted) {
  const int e = blockIdx.x * blockDim.x + threadIdx.x;
  if (e >= N_EDGES) return;
  const int d = dst[e];
  const int pos = seg_start[d] + atomicAdd(&cursor[d], 1);
  eid_sorted[pos] = e;
  dist_sorted[pos] = dist[e];
}

// ---------------------------------------------------------------------------
// Kernel 6: per-node top-k selection (k = ceil(deg/2)) by O(deg^2) ranking.
// Comparator (dist, eid) reproduces lexsort's stable tie-break exactly and is
// a strict total order -> rank is unique -> sel_eid[start+rank] is race-free
// and deterministic. Selected eids land in rank order -> deterministic sums.
// ---------------------------------------------------------------------------
__global__ void select_kernel(const int* __restrict__ seg_start,
                              const int* __restrict__ deg,
                              const int* __restrict__ eid_sorted,
                              const float* __restrict__ dist_sorted,
                              int* __restrict__ sel_eid) {
  const int nd = blockIdx.x * blockDim.x + threadIdx.x;
  if (nd >= N_NODES) return;
  const int start = seg_start[nd];
  const int dg = deg[nd];
  const int k = (dg + 1) >> 1;  // ceil(0.5 * deg)
  for (int i = 0; i < dg; ++i) {
    const float di = dist_sorted[start + i];
    const int ei = eid_sorted[start + i];
    int rank = 0;
    for (int j = 0; j < dg; ++j) {
      const float dj = dist_sorted[start + j];
      const int ej = eid_sorted[start + j];
      rank += (dj < di) || (dj == di && ej < ei);
    }
    if (rank < k) sel_eid[start + rank] = ei;
  }
}

// ---------------------------------------------------------------------------
// Kernel 7: scatter-mean over selected edges + residual:
// h = feat + (k>0 ? sum(feat[src])/k : feat). One wave per node,
// 1 float4 per lane (32*16B = 512 B rows, coalesced; feat is L2-resident).
// ---------------------------------------------------------------------------
__global__ __launch_bounds__(256) void aggregate_kernel(
    const float* __restrict__ feat, const int* __restrict__ src,
    const int* __restrict__ seg_start, const int* __restrict__ deg,
    const int* __restrict__ sel_eid, float* __restrict__ h) {
  const int node = blockIdx.x * 8 + (threadIdx.x >> 5);
  const int lane = threadIdx.x & 31;
  if (node >= N_NODES) return;
  const int start = seg_start[node];
  const int dg = deg[node];
  const int k = (dg + 1) >> 1;

  float4 acc = make_float4(0.f, 0.f, 0.f, 0.f);
  for (int i = 0; i < k; ++i) {
    const int s = src[sel_eid[start + i]];
    if (i + 1 < k)
      __builtin_prefetch(feat + (size_t)src[sel_eid[start + i + 1]] * IN_DIM, 0, 1);
    const float4 v = *((const float4*)(feat + (size_t)s * IN_DIM) + lane);
    acc.x += v.x; acc.y += v.y; acc.z += v.z; acc.w += v.w;
  }
  const float4 f = *((const float4*)(feat + (size_t)node * IN_DIM) + lane);
  float4 hr;
  if (k > 0) {
    const float inv = 1.0f / (float)k;
    hr = make_float4(acc.x * inv, acc.y * inv, acc.z * inv, acc.w * inv);
  } else {
    hr = f;
  }
  float4 o = make_float4(f.x + hr.x, f.y + hr.y, f.z + hr.z, f.w + hr.w);
  *((float4*)(h + (size_t)node * IN_DIM) + lane) = o;
}

// ---------------------------------------------------------------------------
// Kernel 8: out = h @ W_lin + b_lin   [50000,128]x[128,128]
// WMMA f32 16x16x4, 8 waves per block = 8 column tiles, block = 16-row strip.
// W_lin (64 KB) is staged into LDS once per block via GLOBAL_LOAD_ASYNC_TO_LDS
// (ASYNCcnt-tracked, no VGPR round trip), drained with s_wait_asynccnt, then
// the WMMA B operands come from ds_load. Generic->LDS address: low 32 bits of
// a flat pointer into __shared__ are the LDS byte offset (ISA 10.2).
// ---------------------------------------------------------------------------
__global__ __launch_bounds__(256) void gemm_kernel(
    const float* __restrict__ h, const float* __restrict__ W,
    const float* __restrict__ b, float* __restrict__ out) {
  __shared__ float Ws[IN_DIM * OUT_DIM];  // 64 KB

  // Cooperative async stage: 16384 floats = 256 threads x 16 rounds x float4.
  {
    const int tid = threadIdx.x;
#pragma unroll
    for (int r = 0; r < 16; ++r) {
      const int idx4 = (r * 256 + tid) * 4;  // float index, 16B granular
      const unsigned lds_off = (unsigned)(uintptr_t)(&Ws[idx4]);
      const float* gaddr = W + idx4;
      asm volatile("global_load_async_to_lds_b128 %0, %1, off"
                   :: "v"(lds_off), "v"(gaddr)
                   : "memory");
    }
    asm volatile("s_wait_asynccnt 0x0" ::: "memory");
  }
  __syncthreads();

  const int lane = threadIdx.x & 31;
  const int wave = threadIdx.x >> 5;
  const int hf = lane >> 4;
  const int m = lane & 15;
  const int row0 = blockIdx.x * 16;
  const int n0 = wave * 16;

  const float* arow = h + (size_t)(row0 + m) * IN_DIM;
  v8f acc = {};
#pragma unroll 8
  for (int k = 0; k < IN_DIM; k += 4) {
    const int ka = k + 2 * hf;
    v2f a;
    a.x = arow[ka];
    a.y = arow[ka + 1];
    v2f bb;
    bb.x = Ws[ka * OUT_DIM + n0 + m];
    bb.y = Ws[(ka + 1) * OUT_DIM + n0 + m];
    acc = __builtin_amdgcn_wmma_f32_16x16x4_f32(
        false, a, false, bb, (short)0, acc, false, false);
  }
  const float bias = b[n0 + m];
#pragma unroll
  for (int v = 0; v < 8; ++v) {
    out[(size_t)(row0 + v + 8 * hf) * OUT_DIM + n0 + m] = acc[v] + bias;
  }
}

// ---------------------------------------------------------------------------
extern "C" void kernel_launch(void* const* d_in, const int* in_sizes, int n_in,
                              void* d_out, int out_size, void* d_ws, size_t ws_size,
                              hipStream_t stream) {
  const float* feat = (const float*)d_in[0];
  const int*   src  = (const int*)d_in[1];
  const int*   dst  = (const int*)d_in[2];
  const float* Wm   = (const float*)d_in[3];
  const float* bm   = (const float*)d_in[4];
  const float* Wl   = (const float*)d_in[5];
  const float* bl   = (const float*)d_in[6];
  float* out = (float*)d_out;

  // Workspace carve-up (all chunks 16B-aligned): ~46.9 MB total
  char* w = (char*)d_ws;
  float* t           = (float*)w; w += (size_t)N_NODES * N_CLS * 4;      // 8.0 MB
  float* dist        = (float*)w; w += (size_t)N_EDGES * 4;              // 3.2 MB
  float* dist_sorted = (float*)w; w += (size_t)N_EDGES * 4;              // 3.2 MB
  int*   eid_sorted  = (int*)w;   w += (size_t)N_EDGES * 4;              // 3.2 MB
  int*   sel_eid     = (int*)w;   w += (size_t)N_EDGES * 4;              // 3.2 MB
  int*   deg         = (int*)w;   w += (size_t)N_NODES * 4;              // 0.2 MB
  int*   seg_start   = (int*)w;   w += (size_t)N_NODES * 4;              // 0.2 MB
  int*   cursor      = (int*)w;   w += (size_t)N_NODES * 4;              // 0.2 MB
  float* Wpad        = (float*)w; w += (size_t)IN_DIM * N_CLS_PAD * 4;   // 24 KB
  float* h           = (float*)w; w += (size_t)N_NODES * IN_DIM * 4;     // 25.6 MB

  hipMemsetAsync(deg, 0, (size_t)N_NODES * 4, stream);
  hipMemsetAsync(cursor, 0, (size_t)N_NODES * 4, stream);

  padW_kernel<<<(IN_DIM * N_CLS_PAD + 255) / 256, 256, 0, stream>>>(Wm, Wpad);
  mlp_tanh_kernel<<<N_NODES / 16, 32, 0, stream>>>(feat, Wpad, bm, t);
  dist_kernel<<<(N_EDGES + 255) / 256, 256, 0, stream>>>(src, dst, t, dist);
  degree_kernel<<<(N_EDGES + 255) / 256, 256, 0, stream>>>(dst, deg);
  scan_kernel<<<1, 1024, 0, stream>>>(deg, seg_start, N_NODES);
  scatter_kernel<<<(N_EDGES + 255) / 256, 256, 0, stream>>>(
      dst, dist, seg_start, cursor, eid_sorted, dist_sorted);
  select_kernel<<<(N_NODES + 255) / 256, 256, 0, stream>>>(
      seg_start, deg, eid_sorted, dist_sorted, sel_eid);
  aggregate_kernel<<<(N_NODES + 7) / 8, 256, 0, stream>>>(
      feat, src, seg_start, deg, sel_eid, h);
  gemm_kernel<<<N_NODES / 16, 256, 0, stream>>>(h, Wl, bl, out);
}